// MaskingSlidingWindow_34333968564594
// MI455X (gfx1250) — compile-verified
//
#include <hip/hip_runtime.h>

typedef __attribute__((ext_vector_type(2))) float v2f;
typedef __attribute__((ext_vector_type(8))) float v8f;

#define B_DIM 16
#define S_DIM 2048
#define E_DIM 128
#define W_DIM 128
#define V_DIM 7
#define FCW_ROW (E_DIM * W_DIM)   // 16384 floats per fc_w row
#define LDS_STRIDE 132            // 128 + 4 pad -> conflict-free column reads

// -------- d_ws layout (bytes) --------
// [0      , 65536): w_eff   : 16384 f32   (folded conv weights, index e*128+k)
// [65536  , 73728): M       : 16x128 f32  (emb @ w_eff; rows 7..15 == 0)
// [73728  , 81920): win     : 16x128 i32  (window tokens, sentinel 7 = OOB)

// Kernel 1: per batch, count non-pad tokens -> last index, then extract the
// 128 window tokens ending at last (sentinel 7 for positions < 0).
__global__ void k1_window(const int* __restrict__ x, int* __restrict__ win) {
    __shared__ int red[256];
    const int b   = blockIdx.x;
    const int tid = threadIdx.x;
    const int* xb = x + b * S_DIM;
    int cnt = 0;
    for (int i = tid; i < S_DIM; i += 256) cnt += (xb[i] != 0) ? 1 : 0;
    red[tid] = cnt;
    __syncthreads();
    for (int s = 128; s > 0; s >>= 1) {
        if (tid < s) red[tid] += red[tid + s];
        __syncthreads();
    }
    const int last = red[0] - 1;
    if (tid < W_DIM) {
        const int p = last - (W_DIM - 1) + tid;
        int tok = (p >= 0 && p < S_DIM) ? xb[p] : V_DIM;
        if (tok < 0 || tok >= V_DIM) tok = V_DIM;   // sentinel -> zero row of M
        win[b * W_DIM + tid] = tok;
    }
}

// Kernel 2: w_eff[j] = sum_o out_w[o] * fc_w[o*16384 + j].  One streaming,
// fully-coalesced pass over the 8 MB fc_w (the roofline term).  float4 per
// thread; global_prefetch_b8 pulls the row 8 iterations ahead into cache.
__global__ void k2_weff(const float* __restrict__ fc_w,
                        const float* __restrict__ out_w,
                        float* __restrict__ w_eff) {
    const int j = (blockIdx.x * blockDim.x + threadIdx.x) * 4;  // 0..16380
    float4 acc = make_float4(0.f, 0.f, 0.f, 0.f);
    for (int o = 0; o < E_DIM; ++o) {
        int op = o + 8; op = (op > E_DIM - 1) ? (E_DIM - 1) : op;   // branchless clamp
        __builtin_prefetch(fc_w + (size_t)op * FCW_ROW + j, 0, 1);
        const float4 r = *reinterpret_cast<const float4*>(fc_w + (size_t)o * FCW_ROW + j);
        const float wo = out_w[o];               // uniform -> scalar load
        acc.x += wo * r.x; acc.y += wo * r.y;
        acc.z += wo * r.z; acc.w += wo * r.w;
    }
    *reinterpret_cast<float4*>(w_eff + j) = acc;
}

// Kernel 3: M(16x128) = Apad(16x128) x Wmat(128x128) with V_WMMA_F32_16X16X4_F32.
// Apad rows 0..6 = emb_table, rows 7..15 = 0, staged once into LDS so the
// WMMA loop is branch-free (EXEC stays all-ones throughout).
// One block, 8 waves; wave w owns N-tile k in [16w, 16w+16), 32 K-steps of 4.
__global__ void k3_wmma(const float* __restrict__ emb,
                        const float* __restrict__ w_eff,
                        float* __restrict__ Mout) {
    __shared__ float As[16 * LDS_STRIDE];

    const int tid = threadIdx.x;
    // cooperative zero-padded stage of A (16x128 logical, stride-132 in LDS)
    for (int i = tid; i < 16 * E_DIM; i += 256) {
        const int r = i >> 7;          // 0..15
        const int c = i & (E_DIM - 1); // 0..127
        As[r * LDS_STRIDE + c] = (r < V_DIM) ? emb[r * E_DIM + c] : 0.f;
    }
    __syncthreads();

    const int lane = tid & 31;
    const int wave = tid >> 5;           // 0..7 -> N tile
    const int half = lane >> 4;          // K-pair select within fragment
    const int lrow = lane & 15;          // A row (M) / B column within tile
    const int ncol = wave * 16 + lrow;   // global N column
    const float* arow = As + lrow * LDS_STRIDE;

    v8f c = {};
    for (int ki = 0; ki < 32; ++ki) {
        const int kb = ki * 4 + half * 2;          // e (= K) base for this lane
        v2f a, bf;
        a.x = arow[kb];                            // ds_load_b64 (8B aligned pair)
        a.y = arow[kb + 1];
        bf.x = w_eff[(kb)     * W_DIM + ncol];
        bf.y = w_eff[(kb + 1) * W_DIM + ncol];
        // 8 args: (neg_a, A, neg_b, B, c_mod, C, reuse_a, reuse_b)
        c = __builtin_amdgcn_wmma_f32_16x16x4_f32(
                false, a, false, bf, (short)0, c, false, false);
    }
    // D layout: VGPR i -> row i (lanes 0-15) / row i+8 (lanes 16-31)
#pragma unroll
    for (int i = 0; i < 8; ++i) {
        const int row = half * 8 + i;
        Mout[row * W_DIM + ncol] = c[i];
    }
}

// Kernel 4: out[b] = out_b + sum_k ( out_w[k]*fc_b[k] + M[win[b,k], k] )
__global__ void k4_out(const int* __restrict__ win, const float* __restrict__ M,
                       const float* __restrict__ fc_b, const float* __restrict__ out_w,
                       const float* __restrict__ out_b, float* __restrict__ out) {
    __shared__ float red[128];
    const int b = blockIdx.x;
    const int k = threadIdx.x;           // 0..127
    const int tok = win[b * W_DIM + k];  // 0..7 ; row 7 of M is zero
    red[k] = M[tok * W_DIM + k] + out_w[k] * fc_b[k];
    __syncthreads();
    for (int s = 64; s > 0; s >>= 1) {
        if (k < s) red[k] += red[k + s];
        __syncthreads();
    }
    if (k == 0) out[b] = red[0] + out_b[0];
}

extern "C" void kernel_launch(void* const* d_in, const int* in_sizes, int n_in,
                              void* d_out, int out_size, void* d_ws, size_t ws_size,
                              hipStream_t stream) {
    const int*   x     = (const int*)  d_in[0];   // [16, 2048]
    const float* emb   = (const float*)d_in[1];   // [7, 128]
    const float* fc_w  = (const float*)d_in[2];   // [128, 16384]
    const float* fc_b  = (const float*)d_in[3];   // [128]
    const float* out_w = (const float*)d_in[4];   // [1, 128]
    const float* out_b = (const float*)d_in[5];   // [1]
    float* out = (float*)d_out;                   // [16]

    char* ws     = (char*)d_ws;
    float* w_eff = (float*)(ws);            // 65536 B
    float* Mmat  = (float*)(ws + 65536);    //  8192 B
    int*   win   = (int*)  (ws + 73728);    //  8192 B

    k1_window<<<B_DIM, 256, 0, stream>>>(x, win);
    k2_weff  <<<FCW_ROW / (4 * 256), 256, 0, stream>>>(fc_w, out_w, w_eff);
    k3_wmma  <<<1, 256, 0, stream>>>(emb, w_eff, Mmat);
    k4_out   <<<B_DIM, W_DIM, 0, stream>>>(win, Mmat, fc_b, out_w, out_b, out);
}